// TensorBSplinesClassifier_28312424415393
// MI455X (gfx1250) — compile-verified
//
#include <hip/hip_runtime.h>
#include <math.h>
#include <stdint.h>

typedef __attribute__((ext_vector_type(2))) float v2f;
typedef __attribute__((ext_vector_type(8))) float v8f;

#define DATA_DIM   512
#define BASIS_DIM  64
#define NSPAN      61            // number of interior knot intervals; h = 1/61
#define TILE_B     16            // batch rows per block (= WMMA M)
#define THREADS    256           // 8 wave32s
#define WAVES      8
#define I_PER_WAVE (DATA_DIM / WAVES)   // 64 data dims per wave

// Hardware fast reciprocal (v_rcp_f32, ~1 ulp): denominators are knot
// differences >= h = 1/61, well-conditioned, so IEEE divide is overkill.
__device__ __forceinline__ float frcp(float v) { return __builtin_amdgcn_rcpf(v); }

// Clamped open-uniform knot vector collapses to t(i) = clamp((i-3)*h, 0, 1).
__device__ __forceinline__ float knotv(int i) {
    float v = (float)(i - 3) * (1.0f / (float)NSPAN);
    return fminf(fmaxf(v, 0.0f), 1.0f);
}

// de Boor BasisFuns for degree 3: returns dot(N[0..3], Arow[s..s+3]).
__device__ __forceinline__ float bspline_contrib(float x, const float* __restrict__ Arow) {
    int s = (int)(x * (float)NSPAN);
    s = s < 0 ? 0 : (s > NSPAN - 1 ? NSPAN - 1 : s);   // span: t[s+3] <= x < t[s+4]
    const float l1 = x - knotv(s + 3);
    const float l2 = x - knotv(s + 2);
    const float l3 = x - knotv(s + 1);
    const float r1 = knotv(s + 4) - x;
    const float r2 = knotv(s + 5) - x;
    const float r3 = knotv(s + 6) - x;

    float N0, N1, N2, N3, saved, temp;
    // j = 1
    temp = frcp(r1 + l1);
    N0 = r1 * temp;
    N1 = l1 * temp;
    // j = 2
    temp = N0 * frcp(r1 + l2);
    N0 = r1 * temp;
    saved = l2 * temp;
    temp = N1 * frcp(r2 + l1);
    N1 = saved + r2 * temp;
    N2 = l1 * temp;
    // j = 3
    temp = N0 * frcp(r1 + l3);
    N0 = r1 * temp;
    saved = l3 * temp;
    temp = N1 * frcp(r2 + l2);
    N1 = saved + r2 * temp;
    saved = l2 * temp;
    temp = N2 * frcp(r3 + l1);
    N2 = saved + r3 * temp;
    N3 = l1 * temp;

    return N0 * Arow[s] + N1 * Arow[s + 1] + N2 * Arow[s + 2] + N3 * Arow[s + 3];
}

__global__ __launch_bounds__(THREADS)
void bspline_classifier_kernel(const float* __restrict__ x,
                               const float* __restrict__ A,
                               float* __restrict__ out)
{
    __shared__ float xtile[TILE_B * DATA_DIM];   // 32 KB: 16 batch rows of x
    __shared__ float part[WAVES][TILE_B];        // cross-wave partial sums

    const int tid   = threadIdx.x;
    const int wave  = tid >> 5;
    const int lane  = tid & 31;
    const int brow  = lane & 15;   // M row within tile (WMMA A-operand layout)
    const int khalf = lane >> 4;   // lanes 16-31 own K-slots 2,3

    // ---- Stage 16 contiguous batch rows (contiguous 32 KB of x) into LDS ----
    // CDNA5 async memory->LDS path (ASYNCcnt-tracked), 128 B per thread.
    {
        const uint8_t* gsrc = (const uint8_t*)(x + (size_t)blockIdx.x * TILE_B * DATA_DIM);
        const uint32_t lds_base = (uint32_t)(uintptr_t)&xtile[0];  // addr[31:0] = LDS offset
        #pragma unroll
        for (int v = 0; v < (TILE_B * DATA_DIM * 4) / (16 * THREADS); ++v) {
            const uint32_t off = (uint32_t)(tid + v * THREADS) * 16u;
            const uint64_t ga  = (uint64_t)(uintptr_t)(gsrc + off);
            const uint32_t la  = lds_base + off;
            asm volatile("global_load_async_to_lds_b128 %0, %1, off"
                         :: "v"(la), "v"(ga) : "memory");
        }
#if __has_builtin(__builtin_amdgcn_s_wait_asynccnt)
        __builtin_amdgcn_s_wait_asynccnt(0);
#else
        asm volatile("s_wait_asynccnt 0" ::: "memory");
#endif
    }
    __syncthreads();

    // Each lane accumulates two K-slot partial sums over its 64 data dims.
    float acc0 = 0.0f, acc1 = 0.0f;
    const int ibase = wave * I_PER_WAVE;
    #pragma unroll 4
    for (int it = 0; it < I_PER_WAVE / 4; ++it) {
        const int i0 = ibase + it * 4 + khalf * 2;
        const v2f xv = *(const v2f*)&xtile[brow * DATA_DIM + i0];
        acc0 += bspline_contrib(xv.x, A + (size_t)i0 * BASIS_DIM);
        acc1 += bspline_contrib(xv.y, A + (size_t)(i0 + 1) * BASIS_DIM);
    }

    // One f32 WMMA reduces the 4 K-slots per batch row: D[m,n] = sum_k a[m,k].
    // A-operand 16x4 layout: lane&15 = M; a.x/a.y = K = (lane>>4)*2 + {0,1}.
    v2f afrag; afrag.x = acc0; afrag.y = acc1;
    v2f bfrag; bfrag.x = 1.0f; bfrag.y = 1.0f;   // ones: pure K-reduction
    v8f cfrag = {};
    cfrag = __builtin_amdgcn_wmma_f32_16x16x4_f32(
        /*neg_a=*/false, afrag, /*neg_b=*/false, bfrag,
        /*c_mod=*/(short)0, cfrag, /*reuse_a=*/false, /*reuse_b=*/false);

    // D layout: VGPR r holds M=r (lanes 0-15) / M=r+8 (lanes 16-31), replicated over N.
    if (lane == 0) {
        #pragma unroll
        for (int r = 0; r < 8; ++r) part[wave][r] = cfrag[r];
    } else if (lane == 16) {
        #pragma unroll
        for (int r = 0; r < 8; ++r) part[wave][8 + r] = cfrag[r];
    }
    __syncthreads();

    // Cross-wave reduce + sigmoid + store (16 outputs per block).
    if (tid < TILE_B) {
        float y = 0.0f;
        #pragma unroll
        for (int w = 0; w < WAVES; ++w) y += part[w][tid];
        out[blockIdx.x * TILE_B + tid] = frcp(1.0f + __expf(-y));
    }
}

extern "C" void kernel_launch(void* const* d_in, const int* in_sizes, int n_in,
                              void* d_out, int out_size, void* d_ws, size_t ws_size,
                              hipStream_t stream) {
    const float* x = (const float*)d_in[0];   // [batch, 512]
    const float* A = (const float*)d_in[1];   // [512, 64]
    float* out = (float*)d_out;               // [batch]
    (void)n_in; (void)d_ws; (void)ws_size; (void)out_size;

    const int batch = in_sizes[0] / DATA_DIM; // 2048
    dim3 grid(batch / TILE_B);                // 128 blocks
    bspline_classifier_kernel<<<grid, THREADS, 0, stream>>>(x, A, out);
}